// TTTLinearMemory_18451179503932
// MI455X (gfx1250) — compile-verified
//
#include <hip/hip_runtime.h>
#include <hip/hip_bf16.h>

// TTT-Linear scan for MI455X (gfx1250), fp32 WMMA path + async LDS staging.
// B=4 H=16 NC=64 CS=64 D=64. One workgroup per (b,h); W,b live in LDS across
// the 64-chunk sequential scan. 3 GEMMs/chunk via V_WMMA_F32_16X16X4_F32.
// xk chunks are double-buffered in LDS via GLOBAL_LOAD_ASYNC_TO_LDS_B128
// (ASYNCcnt), overlapping each chunk's fetch with the previous chunk's compute.

#define TB 4
#define TH 16
#define TNC 64
#define TCS 64
#define TD 64
#define LDSP 68           // LDS row stride in floats (16B-aligned, bank-skewed)
#define CHUNK (TCS * TD)  // 4096 floats

typedef __attribute__((ext_vector_type(2))) float v2f;
typedef __attribute__((ext_vector_type(8))) float v8f;

static __device__ inline v8f v8f_zero() {
  v8f z;
#pragma unroll
  for (int i = 0; i < 8; ++i) z[i] = 0.0f;
  return z;
}

// Issue one chunk's worth (16KB) of async global->LDS copies for this thread.
// 128 threads x 8 x b128 = 4096 floats. LDS layout keeps the LDSP padding.
static __device__ inline void async_stage_chunk(const float* __restrict__ gsrc,
                                                float* sdst, int tid) {
  const unsigned long long sbase = (unsigned long long)(uintptr_t)gsrc;
#pragma unroll
  for (int j = 0; j < 8; ++j) {
    const int i4 = tid + 128 * j;
    const int r  = i4 >> 4;            // 16 float4 per 64-float row
    const int c4 = (i4 & 15) * 4;
    const unsigned lds = (unsigned)(uintptr_t)(&sdst[r * LDSP + c4]);
    const unsigned go  = (unsigned)(i4 * 16);  // byte offset in chunk
    asm volatile("global_load_async_to_lds_b128 %0, %1, %2"
                 :
                 : "v"(lds), "v"(go), "s"(sbase)
                 : "memory");
  }
}

static __device__ inline void wait_async0() {
  asm volatile("s_wait_asynccnt 0x0" ::: "memory");
}

__launch_bounds__(128, 1)
__global__ void ttt_linear_scan_kernel(const float* __restrict__ xk,
                                       const float* __restrict__ xv,
                                       const float* __restrict__ weight,
                                       const float* __restrict__ bias,
                                       const float* __restrict__ gamma,
                                       const float* __restrict__ beta,
                                       const float* __restrict__ theta,
                                       const float* __restrict__ theta_bias,
                                       const float* __restrict__ alpha,
                                       float* __restrict__ out) {
  __shared__ float sW[TD * LDSP];     // fast weight, persistent over scan
  __shared__ float sXkA[TCS * LDSP];  // staged xk chunk (double buffer A)
  __shared__ float sXkB[TCS * LDSP];  // staged xk chunk (double buffer B)
  __shared__ float sZ[TCS * LDSP];    // z, then reused for G = eta*grad_x
  __shared__ float sB[TD];            // fast bias, persistent
  __shared__ float sGm[TD], sBt[TD], sTh[TD], sTok[TCS];
  __shared__ float sTb;

  const int bh   = blockIdx.x;   // 0..63 = b*16 + h
  const int h    = bh & (TH - 1);
  const int tid  = threadIdx.x;  // 0..127
  const int lane = tid & 31;
  const int wave = tid >> 5;

  const size_t base_bh = (size_t)bh * (size_t)(TNC * CHUNK);

  // kick off async staging of chunk 0 into buffer A immediately
  async_stage_chunk(xk + base_bh, sXkA, tid);

  // ---- one-time per-head setup ----
  for (int i = tid; i < TD * TD; i += 128) {
    int r = i >> 6, c = i & 63;
    sW[r * LDSP + c] = weight[h * TD * TD + i];
  }
  if (tid < TD) {
    sB[tid]  = bias[h * TD + tid];
    sGm[tid] = gamma[h * TD + tid];
    sBt[tid] = beta[h * TD + tid];
    sTh[tid] = theta[h * TD + tid];
    float ti = 1.0f / (float)(tid + 1) + alpha[tid];
    sTok[tid] = ti > 0.0f ? ti : 0.0f;
  }
  if (tid == 0) sTb = theta_bias[h];

  // WMMA fragment coordinates (wave32)
  const int m0    = wave * 16;          // this wave's 16-row strip
  const int lr16  = lane & 15;          // lane % 16
  const int khalf = (lane >> 4) * 2;    // 0 or 2  (A/B K sub-offset)
  const int crow  = (lane >> 4) * 8;    // C/D row sub-offset

#pragma unroll 1
  for (int nc = 0; nc < TNC; ++nc) {
    float* sx  = (nc & 1) ? sXkB : sXkA;  // current chunk buffer
    float* sxn = (nc & 1) ? sXkA : sXkB;  // next chunk buffer

    // wait for this wave's async loads of the current buffer, then make all
    // waves' portions (and last iteration's LDS work) visible.
    wait_async0();
    __syncthreads();

    const float* xvc  = xv  + base_bh + (size_t)nc * CHUNK;
    float*       outc = out + base_bh + (size_t)nc * CHUNK;

    // start fetching next chunk behind this chunk's compute
    if (nc + 1 < TNC) {
      async_stage_chunk(xk + base_bh + (size_t)(nc + 1) * CHUNK, sxn, tid);
      __builtin_prefetch(xvc + CHUNK + tid * 32, 0, 1);  // warm xv in L2
    }

    // ---- z = xk @ W  (16x16x4 fp32 WMMA), +bias, -> sZ ----
    {
      v8f acc[4];
#pragma unroll
      for (int t = 0; t < 4; ++t) acc[t] = v8f_zero();
#pragma unroll 4
      for (int k0 = 0; k0 < TD; k0 += 4) {
        v2f a;
        a[0] = sx[(m0 + lr16) * LDSP + k0 + khalf];
        a[1] = sx[(m0 + lr16) * LDSP + k0 + khalf + 1];
#pragma unroll
        for (int t = 0; t < 4; ++t) {
          const int n0 = t * 16;
          v2f bm;
          bm[0] = sW[(k0 + khalf) * LDSP + n0 + lr16];
          bm[1] = sW[(k0 + khalf + 1) * LDSP + n0 + lr16];
          acc[t] = __builtin_amdgcn_wmma_f32_16x16x4_f32(
              false, a, false, bm, (short)0, acc[t], false, false);
        }
      }
#pragma unroll
      for (int t = 0; t < 4; ++t) {
        const int n0 = t * 16;
#pragma unroll
        for (int i = 0; i < 8; ++i) {
          int r = m0 + crow + i;
          int c = n0 + lr16;
          sZ[r * LDSP + c] = acc[t][i] + sB[c];
        }
      }
    }
    __syncthreads();

    // ---- elementwise LN + gradient; thread = (row r, half of columns) ----
    {
      const int r  = tid >> 1;
      const int c0 = (tid & 1) * 32;
      float zv[32], gxh[32];
      float smu = 0.0f, sth = 0.0f;
#pragma unroll
      for (int i = 0; i < 32; ++i) {
        zv[i] = sZ[r * LDSP + c0 + i];
        smu += zv[i];
        sth += sx[r * LDSP + c0 + i] * sTh[c0 + i];
      }
      smu += __shfl_xor(smu, 1);
      sth += __shfl_xor(sth, 1);
      const float mu = smu * (1.0f / 64.0f);
      float svar = 0.0f;
#pragma unroll
      for (int i = 0; i < 32; ++i) { float d = zv[i] - mu; svar += d * d; }
      svar += __shfl_xor(svar, 1);
      const float var  = svar * (1.0f / 64.0f);
      const float stdv = sqrtf(var + 1e-6f);
      const float rstd = 1.0f / stdv;
      // per-row learning rate: eta = tok(r) * sigmoid(xk.theta + tb) / D
      const float lr  = 1.0f / (1.0f + __expf(-(sth + sTb)));
      const float eta = sTok[r] * (lr * (1.0f / 64.0f));

      float s1 = 0.0f, s2 = 0.0f;
#pragma unroll
      for (int i = 0; i < 32; ++i) {
        const int c  = c0 + i;
        const float xh = (zv[i] - mu) * rstd;
        zv[i] = xh;  // now x_hat
        const float y  = sGm[c] * xh + sBt[c];
        const float go = y - (xvc[r * 64 + c] - sx[r * LDSP + c]);
        const float gx = go * sGm[c];
        gxh[i] = gx;
        s1 += gx;
        s2 += gx * xh;
      }
      s1 += __shfl_xor(s1, 1);
      s2 += __shfl_xor(s2, 1);
      const float sc = rstd * (1.0f / 64.0f);
#pragma unroll
      for (int i = 0; i < 32; ++i) {
        const float gradx = (64.0f * gxh[i] - s1 - zv[i] * s2) * sc;
        sZ[r * LDSP + c0 + i] = eta * gradx;  // G = eta * grad_x (aliases sZ)
      }
    }
    __syncthreads();

    // ---- db: column sums of G; update fast bias ----
    if (tid < TD) {
      float db = 0.0f;
#pragma unroll 8
      for (int k = 0; k < TCS; ++k) db += sZ[k * LDSP + tid];
      sB[tid] -= db;
    }

    // ---- dW = xk^T @ G ; W -= dW  (A is transposed via LDS addressing) ----
    {
      v8f acc[4];
#pragma unroll
      for (int t = 0; t < 4; ++t) acc[t] = v8f_zero();
#pragma unroll 4
      for (int k0 = 0; k0 < TCS; k0 += 4) {
        v2f a;  // A[d][k] = xk[k][d]
        a[0] = sx[(k0 + khalf) * LDSP + m0 + lr16];
        a[1] = sx[(k0 + khalf + 1) * LDSP + m0 + lr16];
#pragma unroll
        for (int t = 0; t < 4; ++t) {
          const int n0 = t * 16;
          v2f bm;  // B[k][e] = G[k][e]
          bm[0] = sZ[(k0 + khalf) * LDSP + n0 + lr16];
          bm[1] = sZ[(k0 + khalf + 1) * LDSP + n0 + lr16];
          acc[t] = __builtin_amdgcn_wmma_f32_16x16x4_f32(
              false, a, false, bm, (short)0, acc[t], false, false);
        }
      }
#pragma unroll
      for (int t = 0; t < 4; ++t) {
        const int n0 = t * 16;
#pragma unroll
        for (int i = 0; i < 8; ++i) {
          int r = m0 + crow + i;
          int c = n0 + lr16;
          sW[r * LDSP + c] -= acc[t][i];
        }
      }
    }
    __syncthreads();

    // ---- out = xk @ Wn + bn -> global ----
    {
      v8f acc[4];
#pragma unroll
      for (int t = 0; t < 4; ++t) acc[t] = v8f_zero();
#pragma unroll 4
      for (int k0 = 0; k0 < TD; k0 += 4) {
        v2f a;
        a[0] = sx[(m0 + lr16) * LDSP + k0 + khalf];
        a[1] = sx[(m0 + lr16) * LDSP + k0 + khalf + 1];
#pragma unroll
        for (int t = 0; t < 4; ++t) {
          const int n0 = t * 16;
          v2f bm;
          bm[0] = sW[(k0 + khalf) * LDSP + n0 + lr16];
          bm[1] = sW[(k0 + khalf + 1) * LDSP + n0 + lr16];
          acc[t] = __builtin_amdgcn_wmma_f32_16x16x4_f32(
              false, a, false, bm, (short)0, acc[t], false, false);
        }
      }
#pragma unroll
      for (int t = 0; t < 4; ++t) {
        const int n0 = t * 16;
#pragma unroll
        for (int i = 0; i < 8; ++i) {
          int r = m0 + crow + i;
          int c = n0 + lr16;
          outc[r * 64 + c] = acc[t][i] + sB[c];
        }
      }
    }
    // next iteration's top-of-loop barrier orders the buffer swap
  }
}

extern "C" void kernel_launch(void* const* d_in, const int* in_sizes, int n_in,
                              void* d_out, int out_size, void* d_ws, size_t ws_size,
                              hipStream_t stream) {
  const float* xk         = (const float*)d_in[0];
  const float* xv         = (const float*)d_in[1];
  const float* weight     = (const float*)d_in[2];
  const float* bias       = (const float*)d_in[3];
  const float* gamma      = (const float*)d_in[4];
  const float* beta       = (const float*)d_in[5];
  const float* theta      = (const float*)d_in[6];
  const float* theta_bias = (const float*)d_in[7];
  const float* alpha      = (const float*)d_in[8];
  float* outp             = (float*)d_out;

  dim3 grid(TB * TH);   // 64 workgroups, one per (b,h)
  dim3 block(128);      // 4 wave32 waves
  ttt_linear_scan_kernel<<<grid, block, 0, stream>>>(
      xk, xv, weight, bias, gamma, beta, theta, theta_bias, alpha, outp);
}